// iBlockRaw_90409061581007
// MI455X (gfx1250) — compile-verified
//
#include <hip/hip_runtime.h>

// ---------------------------------------------------------------------------
// Fused mini-transformer block for MI455X (gfx1250, wave32).
//   grid  = B (=2048) workgroups, one per sequence of pl=128 tokens
//   block = 128 threads = 4 waves; wave h owns attention head h
//   QK^T       : V_WMMA_F32_16X16X4_F32 (K == QK == 4, exact fit)
//   softmax+PV : VALU + __shfl_xor half-row reduction (wave32)
//   FFN        : both layers on V_WMMA_F32_16X16X4_F32 (K=16 and K=64 chained)
//   FFN weights: TENSOR_LOAD_TO_LDS (TDM DMA) issued at kernel entry,
//                fenced with s_wait_tensorcnt just before the FFN stage
// ---------------------------------------------------------------------------

typedef __attribute__((ext_vector_type(2))) float    v2f;
typedef __attribute__((ext_vector_type(8))) float    v8f;
typedef __attribute__((ext_vector_type(4))) unsigned v4u;
typedef __attribute__((ext_vector_type(8))) unsigned v8u;

#define N_HEAD 4
#define QK 4
#define VD 4
#define PL 128
#define NEG_INF (-1000000000.0f)      // matches reference INF constant
#define THETA1 0.031622776601683794f  // 1000^(-0.5)

// Issue one TDM descriptor: DMA `nelem` DWORDs from `gptr` into LDS offset
// `lds_off`, as a 1-row 2-D tile (D# per CDNA5 ISA §8.3/8.4). Tracked by
// TENSORcnt; EXEC is ignored by TDM, operands must live in SGPRs.
__device__ __forceinline__ void tdm_load_1d(const void* gptr, unsigned lds_off,
                                            unsigned nelem) {
    const unsigned long long ga = (unsigned long long)gptr;
    v4u g0;
    g0[0] = 1u;                                   // count=1, user D#, no gather
    g0[1] = lds_off;                              // lds_addr (bytes)
    g0[2] = (unsigned)(ga & 0xFFFFFFFFu);         // global_addr[31:0]
    g0[3] = (unsigned)((ga >> 32) & 0x01FFFFFFu)  // global_addr[56:32]
            | (2u << 30);                         // type = 2 ("image")
    v8u g1;
    g1[0] = (2u << 16);                           // data_size=2 (4B), mask=0
    g1[1] = (nelem & 0xFFFFu) << 16;              // tensor_dim0[15:0]
    g1[2] = (nelem >> 16) | (1u << 16);           // tensor_dim0[31:16], dim1=1
    g1[3] = (nelem & 0xFFFFu) << 16;              // tile_dim0 = nelem
    g1[4] = 1u;                                   // tile_dim1 = 1, tile_dim2 = 0
    g1[5] = nelem;                                // tensor_dim0_stride[31:0]
    g1[6] = 0u;                                   // stride0[47:32], stride1 lo
    g1[7] = 0u;                                   // stride1 hi
    // 2-operand form (VADDR2/VADDR3 = NULL): tensor of up to 2 dimensions.
    asm volatile("tensor_load_to_lds %0, %1" :: "s"(g0), "s"(g1) : "memory");
}

__global__ __launch_bounds__(128) void fused_block_kernel(
    const float* __restrict__ x,   const float* __restrict__ t,
    const int*   __restrict__ mask,
    const float* __restrict__ Wq,  const float* __restrict__ Wk,
    const float* __restrict__ Wv,  const float* __restrict__ Wout,
    const float* __restrict__ W1,  const float* __restrict__ b1,
    const float* __restrict__ W2,  const float* __restrict__ b2,
    float* __restrict__ out)
{
    __shared__ float xs[PL];
    __shared__ float ts[PL];
    __shared__ int   ms[PL];
    __shared__ float Qs[N_HEAD][PL][QK];       // rope'd, pre-scaled by 1/sqrt(QK)
    __shared__ float Ks[N_HEAD][PL][QK];       // rope'd
    __shared__ float Vs[N_HEAD][PL][VD];
    __shared__ float Ss[N_HEAD][16][PL];       // score slab / FFN hidden scratch
    __shared__ float Os[PL][N_HEAD * VD];      // attention output per token
    __shared__ float W1s[64 * 16];             // filled by TDM during attention
    __shared__ float W2s[16 * 64];             // filled by TDM during attention
    __shared__ float b1s[64];
    __shared__ float b2s[16], Wouts[16], Wqs[16], Wks[16], Wvs[16];

    const int tid  = threadIdx.x;
    const int lane = tid & 31;
    const int h    = tid >> 5;          // wave id == head id
    const int seq  = blockIdx.x;

    // ---- Stage 0a: kick off TDM DMA of the FFN weight matrices -------------
    // Generic __shared__ pointers carry the LDS byte offset in their low bits.
    if (h == 0) {
        tdm_load_1d(W1, (unsigned)(unsigned long long)&W1s[0], 64 * 16);
        tdm_load_1d(W2, (unsigned)(unsigned long long)&W2s[0], 16 * 64);
    }

    // ---- Stage 0b: stage inputs + small weights into LDS --------------------
    if (tid < PL) {
        xs[tid] = x[seq * PL + tid];
        ts[tid] = t[seq * PL + tid];
        ms[tid] = mask[seq * PL + tid];
    }
    if (tid < 64) b1s[tid] = b1[tid];
    if (tid < 16) {
        b2s[tid]   = b2[tid];
        Wouts[tid] = Wout[tid];
        Wqs[tid]   = Wq[tid];
        Wks[tid]   = Wk[tid];
        Wvs[tid]   = Wv[tid];
    }
    __syncthreads();

    // ---- Stage 1: q/k/v = x * W (rank-1) + RoPE, per head -------------------
    {
        const float wq0 = Wqs[h*4+0], wq1 = Wqs[h*4+1], wq2 = Wqs[h*4+2], wq3 = Wqs[h*4+3];
        const float wk0 = Wks[h*4+0], wk1 = Wks[h*4+1], wk2 = Wks[h*4+2], wk3 = Wks[h*4+3];
        const float wv0 = Wvs[h*4+0], wv1 = Wvs[h*4+1], wv2 = Wvs[h*4+2], wv3 = Wvs[h*4+3];
        #pragma unroll
        for (int i = 0; i < PL / 32; ++i) {
            const int   l  = lane + 32 * i;
            const float xv = xs[l];
            const float tv = ts[l];
            const float c0 = cosf(tv),          s0 = sinf(tv);
            const float f1 = tv * THETA1;
            const float c1 = cosf(f1),          s1 = sinf(f1);
            const float q0 = xv * wq0, q1 = xv * wq1, q2 = xv * wq2, q3 = xv * wq3;
            const float k0 = xv * wk0, k1 = xv * wk1, k2 = xv * wk2, k3 = xv * wk3;
            Qs[h][l][0] = (q0 * c0 - q1 * s0) * 0.5f;   // pre-scaled 1/sqrt(QK)
            Qs[h][l][1] = (q0 * s0 + q1 * c0) * 0.5f;
            Qs[h][l][2] = (q2 * c1 - q3 * s1) * 0.5f;
            Qs[h][l][3] = (q2 * s1 + q3 * c1) * 0.5f;
            Ks[h][l][0] = k0 * c0 - k1 * s0;
            Ks[h][l][1] = k0 * s0 + k1 * c0;
            Ks[h][l][2] = k2 * c1 - k3 * s1;
            Ks[h][l][3] = k2 * s1 + k3 * c1;
            Vs[h][l][0] = xv * wv0;
            Vs[h][l][1] = xv * wv1;
            Vs[h][l][2] = xv * wv2;
            Vs[h][l][3] = xv * wv3;
        }
    }
    __syncthreads();

    // WMMA f32 16x16x4 operand layouts (wave32):
    //   A (16x4):  lanes 0-15 -> K=0,1 ; lanes 16-31 -> K=2,3 ; M = lane%16
    //   B (4x16):  lanes 0-15 -> K=0,1 ; lanes 16-31 -> K=2,3 ; N = lane%16
    //   C (16x16): vgpr r -> row r + 8*(lane/16), col = lane%16
    const int m     = lane & 15;
    const int kk    = (lane >> 4) << 1;   // 0 or 2
    const int rbase = (lane >> 4) * 8;

    // ---- Stage 2: attention, one 16-query block at a time -------------------
    for (int qi = 0; qi < PL / 16; ++qi) {
        v2f A;
        A.x = Qs[h][qi * 16 + m][kk];
        A.y = Qs[h][qi * 16 + m][kk + 1];

        #pragma unroll
        for (int kj = 0; kj < PL / 16; ++kj) {
            v2f B;
            B.x = Ks[h][kj * 16 + m][kk];
            B.y = Ks[h][kj * 16 + m][kk + 1];
            v8f c = {0.f, 0.f, 0.f, 0.f, 0.f, 0.f, 0.f, 0.f};
            c = __builtin_amdgcn_wmma_f32_16x16x4_f32(
                    false, A, false, B, (short)0, c, false, false);
            #pragma unroll
            for (int r = 0; r < 8; ++r)
                Ss[h][rbase + r][kj * 16 + m] = c[r];
        }
        __syncthreads();

        // masked softmax over 128 keys + P@V, fused.
        // lanes r and r+16 split the 128 columns of query-row r.
        const int row   = lane & 15;
        const int half  = lane >> 4;
        const int ql    = qi * 16 + row;
        const int mq    = ms[ql];
        const int cbase = half * 64;

        float mx = NEG_INF;
        for (int cc = 0; cc < 64; ++cc) {
            const int col = cbase + cc;
            const float sv = (mq != 0 && ms[col] != 0) ? Ss[h][row][col] : NEG_INF;
            mx = fmaxf(mx, sv);
        }
        mx = fmaxf(mx, __shfl_xor(mx, 16));

        float sum = 0.f, a0 = 0.f, a1 = 0.f, a2 = 0.f, a3 = 0.f;
        for (int cc = 0; cc < 64; ++cc) {
            const int col = cbase + cc;
            const float sv = (mq != 0 && ms[col] != 0) ? Ss[h][row][col] : NEG_INF;
            const float p  = expf(sv - mx);
            sum += p;
            a0 += p * Vs[h][col][0];
            a1 += p * Vs[h][col][1];
            a2 += p * Vs[h][col][2];
            a3 += p * Vs[h][col][3];
        }
        sum += __shfl_xor(sum, 16);
        a0  += __shfl_xor(a0, 16);
        a1  += __shfl_xor(a1, 16);
        a2  += __shfl_xor(a2, 16);
        a3  += __shfl_xor(a3, 16);

        if (half == 0) {
            const float inv = 1.0f / sum;
            Os[ql][h * 4 + 0] = a0 * inv;
            Os[ql][h * 4 + 1] = a1 * inv;
            Os[ql][h * 4 + 2] = a2 * inv;
            Os[ql][h * 4 + 3] = a3 * inv;
        }
        __syncthreads();   // also fences Ss reuse for the next query block
    }

    // ---- TDM fence: FFN weights must be resident in LDS ---------------------
    // Non-issuing waves have TENSORcnt == 0 and fall through immediately.
    __builtin_amdgcn_s_wait_tensorcnt(0);
    __syncthreads();

    // ---- Stage 3: FFN on WMMA ----------------------------------------------
    // Wave h handles token tiles ti = h and h+4 (2 uniform iterations/wave).
    // Layer 1: H(16x64) = O_tile(16x16) @ W1^T(16x64) : 4 blocks x 4 K-chunks
    // Layer 2: F(16x16) = H(16x64) @ W2^T(64x16)      : 16 chained K-chunks
    // Output : y = sum_d (O + F + b2)[d] * Wout[d], reduced with shfl_xor.
    // H is staged in the (now free) per-head score slab Ss[h][0:16][0:64].
    for (int ti = h; ti < 8; ti += 4) {
        // ---- Layer 1 + bias + ReLU ----
        #pragma unroll
        for (int ib = 0; ib < 4; ++ib) {
            v8f hc = {0.f, 0.f, 0.f, 0.f, 0.f, 0.f, 0.f, 0.f};
            #pragma unroll
            for (int kc = 0; kc < 4; ++kc) {
                v2f A, B;
                A.x = Os[ti * 16 + m][kc * 4 + kk];           // token row m, K-dim
                A.y = Os[ti * 16 + m][kc * 4 + kk + 1];
                B.x = W1s[(ib * 16 + m) * 16 + kc * 4 + kk];  // W1^T[k][i], col i
                B.y = W1s[(ib * 16 + m) * 16 + kc * 4 + kk + 1];
                hc = __builtin_amdgcn_wmma_f32_16x16x4_f32(
                         false, A, false, B, (short)0, hc, false, false);
            }
            const float bias = b1s[ib * 16 + m];
            #pragma unroll
            for (int r = 0; r < 8; ++r)
                Ss[h][rbase + r][ib * 16 + m] = fmaxf(hc[r] + bias, 0.0f);
        }
        __syncthreads();

        // ---- Layer 2 ----
        v8f fc = {0.f, 0.f, 0.f, 0.f, 0.f, 0.f, 0.f, 0.f};
        #pragma unroll
        for (int kc = 0; kc < 16; ++kc) {
            v2f A, B;
            A.x = Ss[h][m][kc * 4 + kk];               // H: token row m, hidden K
            A.y = Ss[h][m][kc * 4 + kk + 1];
            B.x = W2s[m * 64 + kc * 4 + kk];           // W2^T[k][d], col d = m
            B.y = W2s[m * 64 + kc * 4 + kk + 1];
            fc = __builtin_amdgcn_wmma_f32_16x16x4_f32(
                     false, A, false, B, (short)0, fc, false, false);
        }

        // ---- residual + Wout projection on C-layout registers ----
        const float bb2 = b2s[m];
        const float wo  = Wouts[m];
        #pragma unroll
        for (int r = 0; r < 8; ++r) {
            const int l = ti * 16 + rbase + r;                  // token index
            float v = (Os[l][m] + fc[r] + bb2) * wo;            // per-dim contrib
            v += __shfl_xor(v, 1);                              // reduce 16 dims
            v += __shfl_xor(v, 2);
            v += __shfl_xor(v, 4);
            v += __shfl_xor(v, 8);
            if (m == r) out[seq * PL + l] = v;
        }
        __syncthreads();   // fence Ss[h] reuse by next tile iteration
    }
}

extern "C" void kernel_launch(void* const* d_in, const int* in_sizes, int n_in,
                              void* d_out, int out_size, void* d_ws, size_t ws_size,
                              hipStream_t stream) {
    const float* x    = (const float*)d_in[0];
    const float* t    = (const float*)d_in[1];
    const int*   mask = (const int*)  d_in[2];
    const float* Wq   = (const float*)d_in[3];
    const float* Wk   = (const float*)d_in[4];
    const float* Wv   = (const float*)d_in[5];
    const float* Wout = (const float*)d_in[6];
    const float* W1   = (const float*)d_in[7];
    const float* b1   = (const float*)d_in[8];
    const float* W2   = (const float*)d_in[9];
    const float* b2   = (const float*)d_in[10];
    float* out = (float*)d_out;

    const int nseq = in_sizes[0] / PL;   // bn*pn = 2048
    fused_block_kernel<<<dim3(nseq), dim3(128), 0, stream>>>(
        x, t, mask, Wq, Wk, Wv, Wout, W1, b1, W2, b2, out);
}